// WaveletTransform2D_34866544509232
// MI455X (gfx1250) — compile-verified
//
#include <hip/hip_runtime.h>
#include <cstdint>

// 2D Haar wavelet: x (BC=1024, H=512, W=512) f32 -> 4 subbands (BC, 256, 256) f32,
// concatenated [LL | LH | HL | HH] in d_out.
//
// Memory-bound (AI ~ 0.5 flop/byte). Data path:
//   global --(global_load_async_to_lds_b128, ASYNCcnt)--> LDS --(ds_load_b128)--> VGPR
//   -> VALU butterfly -> non-temporal global_store_b128.

typedef float v4f __attribute__((ext_vector_type(4)));

#define HAAR_W   512
#define HAAR_H   512
#define HAAR_WO  256
#define HAAR_HO  256
#define TPB      256

__global__ __launch_bounds__(TPB) void haar2d_fwd_kernel(
    const float* __restrict__ x, float* __restrict__ out,
    long long total_threads, long long n4)
{
    // 16 KB LDS staging: 4 x b128 per thread.
    __shared__ v4f tile[TPB * 4];

    const int tid = threadIdx.x;
    const long long t = (long long)blockIdx.x * TPB + tid;
    if (t >= total_threads) return;

    // t -> (bc, h', wq) ; each thread covers output columns [4*wq, 4*wq+4)
    const int wq = (int)(t & 63);            // 64 quad-columns per output row
    const int hp = (int)((t >> 6) & 255);    // output row 0..255
    const long long bc = t >> 14;            // batch*channel index 0..1023

    // Input: rows 2*hp and 2*hp+1, starting column 8*wq (8 input cols -> 4 out pixels)
    const long long row0 = (bc * HAAR_H + 2LL * hp) * HAAR_W + 8LL * wq;
    const float* g0 = x + row0;              // row 2h'
    const float* g1 = g0 + HAAR_W;           // row 2h'+1

    // Wave-relative LDS byte offsets (generic LDS addr low 32 bits == LDS offset).
    const uint32_t l0 = (uint32_t)(uintptr_t)(&tile[(size_t)tid * 4]);
    const uint32_t l1 = l0 + 32u;

    // Async global -> LDS, 16B per lane per instruction (offset applies to both sides).
    asm volatile("global_load_async_to_lds_b128 %0, %1, off"
                 :: "v"(l0), "v"(g0) : "memory");
    asm volatile("global_load_async_to_lds_b128 %0, %1, off offset:16"
                 :: "v"(l0), "v"(g0) : "memory");
    asm volatile("global_load_async_to_lds_b128 %0, %1, off"
                 :: "v"(l1), "v"(g1) : "memory");
    asm volatile("global_load_async_to_lds_b128 %0, %1, off offset:16"
                 :: "v"(l1), "v"(g1) : "memory");
    asm volatile("s_wait_asynccnt 0x0" ::: "memory");

    // Read back own staged data (ds_load_b128).
    const v4f r0a = tile[(size_t)tid * 4 + 0];
    const v4f r0b = tile[(size_t)tid * 4 + 1];
    const v4f r1a = tile[(size_t)tid * 4 + 2];
    const v4f r1b = tile[(size_t)tid * 4 + 3];

    const float e0[8] = {r0a.x, r0a.y, r0a.z, r0a.w, r0b.x, r0b.y, r0b.z, r0b.w};
    const float e1[8] = {r1a.x, r1a.y, r1a.z, r1a.w, r1b.x, r1b.y, r1b.z, r1b.w};

    v4f LL, LH, HL, HH;
#pragma unroll
    for (int j = 0; j < 4; ++j) {
        const float a = e0[2 * j], b = e0[2 * j + 1];
        const float c = e1[2 * j], d = e1[2 * j + 1];
        const float apb = a + b, amb = a - b;
        const float cpd = c + d, cmd = c - d;
        LL[j] = 0.5f * (apb + cpd);   // a+b+c+d
        LH[j] = 0.5f * (amb + cmd);   // a-b+c-d
        HL[j] = 0.5f * (apb - cpd);   // a+b-c-d
        HH[j] = 0.5f * (amb - cmd);   // a-b-c+d
    }

    // Output index within one subband.
    const long long obase = (bc * HAAR_HO + hp) * HAAR_WO + 4LL * wq;

    // Streamed-once results: non-temporal 128-bit stores.
    __builtin_nontemporal_store(LL, (v4f*)(out + obase));
    __builtin_nontemporal_store(LH, (v4f*)(out + n4 + obase));
    __builtin_nontemporal_store(HL, (v4f*)(out + 2 * n4 + obase));
    __builtin_nontemporal_store(HH, (v4f*)(out + 3 * n4 + obase));
}

extern "C" void kernel_launch(void* const* d_in, const int* in_sizes, int n_in,
                              void* d_out, int out_size, void* d_ws, size_t ws_size,
                              hipStream_t stream)
{
    const float* x = (const float*)d_in[0];
    float* out = (float*)d_out;

    const long long n  = (long long)in_sizes[0];          // 16*64*512*512
    const long long bc = n / ((long long)HAAR_W * HAAR_H);// 1024
    const long long n4 = n / 4;                           // elements per subband

    // One thread per 4 output pixels: bc * 256 rows * 64 quad-columns.
    const long long total_threads = bc * HAAR_HO * (HAAR_WO / 4);
    const int blocks = (int)((total_threads + TPB - 1) / TPB);

    haar2d_fwd_kernel<<<blocks, TPB, 0, stream>>>(x, out, total_threads, n4);
}